// SDE_16020228014550
// MI455X (gfx1250) — compile-verified
//
#include <hip/hip_runtime.h>
#include <hip/hip_bf16.h>

// ---------------------------------------------------------------------------
// Lotka-Volterra chemical-Langevin Euler-Maruyama paths on gfx1250 (MI455X).
//
// Memory-bound streaming problem (262 MB @ 23.3 TB/s ~= 11 us floor).
// Strategy: 1 thread = 1 replica; z-noise staged through LDS in 10-step
// chunks with a transposed cooperative pattern so global accesses are
// coalesced, prefetched 3 chunks deep with CDNA5 async global->LDS copies
// tracked by ASYNCcnt. Outputs staged through LDS and stored coalesced.
// ---------------------------------------------------------------------------

#define AS1 __attribute__((address_space(1)))
#define AS3 __attribute__((address_space(3)))

typedef int v2i __attribute__((vector_size(8)));

#if defined(__has_builtin)
#  if __has_builtin(__builtin_amdgcn_global_load_async_to_lds_b64)
#    define HAVE_ASYNC 1
#  endif
#endif
#ifndef HAVE_ASYNC
#  define HAVE_ASYNC 0
#endif

#define NREPS   16384
#define TSTEPS  1000
#define BLK     128              // threads per block == replicas per block
#define TCH     10               // timesteps per LDS chunk
#define NCHUNK  (TSTEPS / TCH)   // 100
#define STRIDE  22               // padded floats per LDS row (payload = 20)
#define NBUF    4                // 3-deep prefetch pipeline + current
#define ITERS   TCH              // 8-byte units per row; BLK units per iter

// Split workgroup barrier that drains DS only (NOT ASYNCcnt), so the async
// prefetch pipeline survives across barriers.
#define WG_BARRIER() \
  asm volatile("s_wait_dscnt 0\n\ts_barrier_signal -1\n\ts_barrier_wait -1" ::: "memory")

#if HAVE_ASYNC
#  define WAIT_ASYNC_LIT(n) asm volatile("s_wait_asynccnt " #n ::: "memory")
#else
#  define WAIT_ASYNC_LIT(n) ((void)0)
#endif

__global__ __launch_bounds__(BLK) void sde_lv_kernel(
    const float* __restrict__ x0, const float* __restrict__ theta,
    const float* __restrict__ z, float* __restrict__ out) {
  __shared__ __align__(16) float zbuf[NBUF][BLK * STRIDE];  // 45056 B
  __shared__ __align__(16) float obuf[BLK * STRIDE];        // 11264 B

  const int tid = threadIdx.x;
  const int r0  = blockIdx.x * BLK;

  const float th1 = theta[(r0 + tid) * 3 + 0];
  const float th2 = theta[(r0 + tid) * 3 + 1];
  const float th3 = theta[(r0 + tid) * 3 + 2];
  float x1 = x0[0];
  float x2 = x0[1];

  const float DT   = 0.1f;
  const float RDT  = 0.31622776601683794f;  // sqrt(0.1)
  const float EPSF = 1e-8f;

  // Cooperatively fetch z[r0:r0+128, c*TCH:(c+1)*TCH, :] into zbuf[c % NBUF].
  // Unit = 8 bytes (one timestep, both components). Up to 10 consecutive lanes cover 80
  // contiguous bytes of one replica row -> coalesced global access.
  auto issue_chunk = [&](int c) {
    float* dstbase = &zbuf[c & (NBUF - 1)][0];
    const int tb = c * TCH;
#pragma unroll
    for (int k = 0; k < ITERS; ++k) {
      int u   = k * BLK + tid;
      int row = u / ITERS;
      int col = u - row * ITERS;
      const float* gsrc = z + (((size_t)(r0 + row) * TSTEPS + tb + col) << 1);
      float*       ldst = dstbase + row * STRIDE + (col << 1);
#if HAVE_ASYNC
      __builtin_amdgcn_global_load_async_to_lds_b64((AS1 v2i*)gsrc,
                                                    (AS3 v2i*)ldst, 0, 0);
#else
      *(float2*)ldst = *(const float2*)gsrc;
#endif
    }
  };

  issue_chunk(0);
  issue_chunk(1);
  issue_chunk(2);

  for (int c = 0; c < NCHUNK; ++c) {
    if (c + 3 < NCHUNK) issue_chunk(c + 3);

    // Wait until chunk c's 10 async ops (this wave's) have landed.
    const int ahead = NCHUNK - 1 - c;
    if (ahead >= 3)      { WAIT_ASYNC_LIT(30); }
    else if (ahead == 2) { WAIT_ASYNC_LIT(20); }
    else if (ahead == 1) { WAIT_ASYNC_LIT(10); }
    else                 { WAIT_ASYNC_LIT(0);  }
    WG_BARRIER();

    // Compute 10 sequential Euler-Maruyama steps from LDS (bank-conflict
    // free: dword bank = 2*(11*lane + t) mod 64, gcd(11,32)==1).
    const float* zin = &zbuf[c & (NBUF - 1)][tid * STRIDE];
    float*       xo  = &obuf[tid * STRIDE];
#pragma unroll
    for (int t = 0; t < TCH; ++t) {
      float2 zz = *(const float2*)(zin + (t << 1));
      float xc1   = fminf(fmaxf(x1, 1e-6f), 1e6f);
      float xc2   = fminf(fmaxf(x2, 1e-6f), 1e6f);
      float inter = th2 * xc1 * xc2;
      float birth = th1 * xc1;
      float death = th3 * xc2;
      float a   = fmaxf(birth + inter, EPSF);
      float l11 = sqrtf(a);
      float l21 = (-inter) / l11;
      float l22 = sqrtf(fmaxf((inter + death) - l21 * l21, EPSF));
      float nx1 = x1 + DT * (birth - inter) + RDT * (l11 * zz.x);
      float nx2 = x2 + DT * (inter - death) + RDT * (l21 * zz.x + l22 * zz.y);
      x1 = fminf(fmaxf(nx1, -1e6f), 1e6f);
      x2 = fminf(fmaxf(nx2, -1e6f), 1e6f);
      float2 xv; xv.x = x1; xv.y = x2;
      *(float2*)(xo + (t << 1)) = xv;
    }
    WG_BARRIER();

    // Cooperative coalesced store of the chunk's outputs.
    const int tb = c * TCH;
#pragma unroll
    for (int k = 0; k < ITERS; ++k) {
      int u   = k * BLK + tid;
      int row = u / ITERS;
      int col = u - row * ITERS;
      float2 v = *(const float2*)&obuf[row * STRIDE + (col << 1)];
      *(float2*)(out + (((size_t)(r0 + row) * TSTEPS + tb + col) << 1)) = v;
    }
    WG_BARRIER();  // obuf reuse guard
  }
}

extern "C" void kernel_launch(void* const* d_in, const int* in_sizes, int n_in,
                              void* d_out, int out_size, void* d_ws, size_t ws_size,
                              hipStream_t stream) {
  const float* x0    = (const float*)d_in[0];
  const float* theta = (const float*)d_in[1];
  const float* z     = (const float*)d_in[2];
  float*       out   = (float*)d_out;
  (void)in_sizes; (void)n_in; (void)out_size; (void)d_ws; (void)ws_size;

  dim3 grid(NREPS / BLK);
  dim3 block(BLK);
  hipLaunchKernelGGL(sde_lv_kernel, grid, block, 0, stream, x0, theta, z, out);
}